// CRF_35347580846837
// MI455X (gfx1250) — compile-verified
//
#include <hip/hip_runtime.h>

typedef __attribute__((ext_vector_type(2))) float v2f;
typedef __attribute__((ext_vector_type(8))) float v8f;

#define NN 50000
#define FF 128
#define EE 800000

// ---------------------------------------------------------------------------
// Edge kernel: one wave32 computes a 16-edge x 16-col tile (4 valid cols).
// D[m][n] = sum_k u[m][k]*W_edge[n][k] + sum_k v[m][k]*W_edge[n][128+k] + b[n]
// A (16x4 f32): lane L<16 -> (M=L, K=0,1), lane L>=16 -> (M=L-16, K=2,3)
// B (4x16 f32): lane L<16 -> (K=0,1, N=L), lane L>=16 -> (K=2,3, N=L-16)
// C/D (16x16 f32): VGPR r, lanes 0-15 -> (M=r,N=lane); 16-31 -> (M=r+8,N=l-16)
// ---------------------------------------------------------------------------
__global__ __launch_bounds__(256) void crf_edge_kernel(
    const float* __restrict__ x,          // [N, 128]
    const long long* __restrict__ ei,     // [2, E] int64
    const float* __restrict__ W_edge,     // [4, 256]
    const float* __restrict__ b_edge,     // [4]
    float* __restrict__ edge_pot)         // [E, 4]
{
    // Zero-pad W_edge to 16 rows in LDS so B columns 4..15 contribute zeros.
    __shared__ float wlds[16 * 256];
    for (int i = threadIdx.x; i < 16 * 256; i += 256) {
        int row = i >> 8;
        int col = i & 255;
        wlds[i] = (row < 4) ? W_edge[row * 256 + col] : 0.0f;
    }
    __syncthreads();

    const int lane = threadIdx.x & 31;
    const int wave = threadIdx.x >> 5;
    const int tile = blockIdx.x * 8 + wave;     // E/16 = 50000 tiles exactly
    const int ebase = tile * 16;

    const int m16 = lane & 15;                  // edge-in-tile (A) / col N (B)
    const int k0  = (lane >> 4) << 1;           // 0 for lanes 0-15, 2 for 16-31

    const long long uidx = ei[(size_t)ebase + m16];
    const long long vidx = ei[(size_t)EE + ebase + m16];

    const float* urow = x + (size_t)uidx * FF + k0;
    const float* vrow = x + (size_t)vidx * FF + k0;
    const int wbase = m16 * 256 + k0;           // LDS column n = m16

    v8f c = {};
#pragma unroll 8
    for (int k = 0; k < 32; ++k) {              // u-half: K = 0..127
        v2f a = *(const v2f*)(urow + 4 * k);
        v2f b = *(const v2f*)(&wlds[wbase + 4 * k]);
        c = __builtin_amdgcn_wmma_f32_16x16x4_f32(false, a, false, b,
                                                  (short)0, c, false, false);
    }
#pragma unroll 8
    for (int k = 0; k < 32; ++k) {              // v-half: K = 128..255
        v2f a = *(const v2f*)(vrow + 4 * k);
        v2f b = *(const v2f*)(&wlds[wbase + FF + 4 * k]);
        c = __builtin_amdgcn_wmma_f32_16x16x4_f32(false, a, false, b,
                                                  (short)0, c, false, false);
    }

    const int n   = m16;                        // output column
    const int hi8 = (lane >> 4) << 3;           // lanes>=16 hold M = r+8
    if (n < 4) {
        const float bias = b_edge[n];
#pragma unroll
        for (int r = 0; r < 8; ++r) {
            edge_pot[(size_t)(ebase + r + hi8) * 4 + n] = c[r] + bias;
        }
    }
}

// ---------------------------------------------------------------------------
// Node kernel: same machinery, 16 nodes x 16 cols (2 valid), K = 128.
// ---------------------------------------------------------------------------
__global__ __launch_bounds__(256) void crf_node_kernel(
    const float* __restrict__ x,          // [N, 128]
    const float* __restrict__ W_node,     // [2, 128]
    const float* __restrict__ b_node,     // [2]
    float* __restrict__ node_pot)         // [N, 2]
{
    __shared__ float wlds[16 * 128];
    for (int i = threadIdx.x; i < 16 * 128; i += 256) {
        int row = i >> 7;
        int col = i & 127;
        wlds[i] = (row < 2) ? W_node[row * 128 + col] : 0.0f;
    }
    __syncthreads();

    const int lane = threadIdx.x & 31;
    const int wave = threadIdx.x >> 5;
    const int tile = blockIdx.x * 8 + wave;
    if (tile >= NN / 16) return;                // wave-uniform guard (3125 tiles)
    const int nbase = tile * 16;

    const int m16 = lane & 15;
    const int k0  = (lane >> 4) << 1;

    const float* arow = x + (size_t)(nbase + m16) * FF + k0;
    const int wbase = m16 * 128 + k0;

    v8f c = {};
#pragma unroll 8
    for (int k = 0; k < 32; ++k) {
        v2f a = *(const v2f*)(arow + 4 * k);
        v2f b = *(const v2f*)(&wlds[wbase + 4 * k]);
        c = __builtin_amdgcn_wmma_f32_16x16x4_f32(false, a, false, b,
                                                  (short)0, c, false, false);
    }

    const int n   = m16;
    const int hi8 = (lane >> 4) << 3;
    if (n < 2) {
        const float bias = b_node[n];
#pragma unroll
        for (int r = 0; r < 8; ++r) {
            node_pot[(size_t)(nbase + r + hi8) * 2 + n] = c[r] + bias;
        }
    }
}

extern "C" void kernel_launch(void* const* d_in, const int* in_sizes, int n_in,
                              void* d_out, int out_size, void* d_ws, size_t ws_size,
                              hipStream_t stream) {
    (void)in_sizes; (void)n_in; (void)out_size; (void)d_ws; (void)ws_size;

    const float*     x      = (const float*)d_in[0];     // [N,128] f32
    const long long* ei     = (const long long*)d_in[1]; // [2,E] int64
    const float*     W_node = (const float*)d_in[2];     // [2,128]
    const float*     b_node = (const float*)d_in[3];     // [2]
    const float*     W_edge = (const float*)d_in[4];     // [4,256]
    const float*     b_edge = (const float*)d_in[5];     // [4]

    float* node_pot = (float*)d_out;                     // N*2 floats
    float* edge_pot = (float*)d_out + (size_t)NN * 2;    // E*4 floats

    // Node: 3125 tiles, 8 waves/block -> 391 blocks (guarded).
    crf_node_kernel<<<(NN / 16 + 7) / 8, 256, 0, stream>>>(x, W_node, b_node, node_pot);
    // Edge: 50000 tiles, 8 waves/block -> 6250 blocks exactly.
    crf_edge_kernel<<<EE / 16 / 8, 256, 0, stream>>>(x, ei, W_edge, b_edge, edge_pot);
}